// clDiceBCELoss_16329465659738
// MI455X (gfx1250) — compile-verified
//
#include <hip/hip_runtime.h>
#include <math.h>

// clDice + BCE + Dice fused loss for B=16, 1x1024x1024 f32 tensors.
// Stage 1: one block per 64x64 tile; both skeletons computed in LDS with a
//          12-px halo (async global->LDS fill); 8 partial sums per block.
// Stage 2: deterministic fixed-order reduction over 4096 block partials.

typedef float v2f __attribute__((ext_vector_type(2)));
typedef float v8f __attribute__((ext_vector_type(8)));

#define IMG   1024
#define TILE  64
#define HALO  12
#define EXT   (TILE + 2 * HALO)   // 88
#define NTHREADS 256

// Async global->LDS copy (CDNA5): per-lane global addr, per-lane LDS addr.
// LDS byte address = low 32 bits of the generic pointer (LDS aperture keeps
// the offset in addr[31:0]); HW adds the wave's LDS_BASE. Tracked by ASYNCcnt.
__device__ __forceinline__ void async_load_to_lds_b32(void* lds_ptr,
                                                      const void* gptr) {
  unsigned int       l = (unsigned int)(unsigned long long)(uintptr_t)lds_ptr;
  unsigned long long g = (unsigned long long)(uintptr_t)gptr;
  asm volatile("global_load_async_to_lds_b32 %0, %1, off"
               :: "v"(l), "v"(g) : "memory");
}
__device__ __forceinline__ void wait_asynccnt0() {
  asm volatile("s_wait_asynccnt 0" ::: "memory");
}

// Full 32-lane f32 sum via V_WMMA_F32_16X16X4_F32.
// A(16x4) with a={x,0}: A[m][0]=x[m], A[m][2]=x[m+16]; B = ones ->
// D[m][n] = x[m]+x[m+16]. Lane n<16 holds rows 0..7, lane>=16 rows 8..15;
// sum of own 8 D vgprs + partner lane (xor 16) = full 32-lane sum.
__device__ __forceinline__ float wave_sum32(float x) {
  v2f a; a[0] = x;    a[1] = 0.0f;
  v2f b; b[0] = 1.0f; b[1] = 1.0f;
  v8f c = {};
  c = __builtin_amdgcn_wmma_f32_16x16x4_f32(false, a, false, b,
                                            (short)0, c, false, false);
  float t = c[0] + c[1] + c[2] + c[3] + c[4] + c[5] + c[6] + c[7];
  t += __shfl_xor(t, 16, 32);
  return t;
}

__device__ __forceinline__ float sigmoidf_fast(float x) {
  return 1.0f / (1.0f + __expf(-x));
}

__global__ __launch_bounds__(NTHREADS)
void cldice_stage1(const float* __restrict__ pred,
                   const float* __restrict__ targ,
                   float* __restrict__ partial) {
  __shared__ float buf0[EXT * EXT];
  __shared__ float buf1[EXT * EXT];
  __shared__ float wred[8 * 8];   // [wave][quantity]

  const int tx = threadIdx.x, ty = threadIdx.y;
  const int tid = ty * 16 + tx;
  const int bx = blockIdx.x * TILE;
  const int by = blockIdx.y * TILE;
  const size_t base = (size_t)blockIdx.z * IMG * IMG;

  float a_bce = 0.f, a_p = 0.f, a_y = 0.f, a_py = 0.f;
  float a_skp = 0.f, a_skpy = 0.f, a_skt = 0.f, a_sktp = 0.f;

  for (int phase = 0; phase < 2; ++phase) {
    float* pA = buf0;
    float* pB = buf1;
    const float* src = (phase == 0) ? pred : targ;

    // ---- async halo fill: 88x88, clamp-to-edge (== 'SAME' pad for min/max)
    for (int idx = tid; idx < EXT * EXT; idx += NTHREADS) {
      int r = idx / EXT - HALO + by;   // const divisor -> mul/shift
      int c = idx % EXT - HALO + bx;
      r = min(max(r, 0), IMG - 1);
      c = min(max(c, 0), IMG - 1);
      size_t g = base + (size_t)r * IMG + c;
      async_load_to_lds_b32(&pA[idx], &src[g]);
      if (phase == 0)
        __builtin_prefetch(&targ[g], 0, 3);   // warm GL2 for phase 1
    }
    wait_asynccnt0();
    if (phase == 0) {
      // In-place sigmoid: each lane transforms exactly the elements it
      // fetched above (same strided index set), so only its own ASYNCcnt
      // wait is required before this loop.
      for (int idx = tid; idx < EXT * EXT; idx += NTHREADS)
        pA[idx] = sigmoidf_fast(pA[idx]);
    }
    __syncthreads();

    // ---- skeleton chain: er_i = erode(er_{i-1}) (5-point min);
    //      i==1: skel = relu(er0 - dilate(er1))            (== relu(x-open(x)))
    //      i>=2: delta_{i-1} = relu(er_{i-1} - dilate(er_i)); skel update.
    // Erode is computed on the fixed region [1..86]^2: ring k of er_i reads
    // rings <= k+1 of er_{i-1}, valid iff k <= 12-i — exactly the needed
    // extent. Stale outer rings never reach the tile center.
    float s[16];
    for (int i = 1; i <= 11; ++i) {
      for (int idx = tid; idx < EXT * EXT; idx += NTHREADS) {
        int r = idx / EXT;
        int c = idx - r * EXT;
        if (r >= 1 && r < EXT - 1 && c >= 1 && c < EXT - 1) {
          float e = fminf(pA[idx],
                    fminf(fminf(pA[idx - 1], pA[idx + 1]),
                          fminf(pA[idx - EXT], pA[idx + EXT])));
          pB[idx] = e;
        }
      }
      __syncthreads();
      // center: dilate (3x3 max) of er_i, combine with er_{i-1}
      for (int k = 0; k < 16; ++k) {
        int rr = ty * 4 + (k >> 2);
        int cc = (k & 3) * 16 + tx;
        int o = (rr + HALO) * EXT + (cc + HALO);
        float d = pB[o];
        d = fmaxf(d, fmaxf(pB[o - 1], pB[o + 1]));
        d = fmaxf(d, fmaxf(pB[o - EXT], pB[o + EXT]));
        d = fmaxf(d, fmaxf(pB[o - EXT - 1], pB[o - EXT + 1]));
        d = fmaxf(d, fmaxf(pB[o + EXT - 1], pB[o + EXT + 1]));
        float delta = fmaxf(pA[o] - d, 0.0f);
        if (i == 1) s[k] = delta;
        else        s[k] += fmaxf(delta - s[k] * delta, 0.0f);
      }
      __syncthreads();
      float* t = pA; pA = pB; pB = t;
    }

    // ---- accumulate partial sums over this tile's 64x64 center
    for (int k = 0; k < 16; ++k) {
      int rr = by + ty * 4 + (k >> 2);
      int cc = bx + (k & 3) * 16 + tx;
      size_t g = base + (size_t)rr * IMG + cc;
      float xv = pred[g];
      float yv = targ[g];
      float pv = sigmoidf_fast(xv);
      if (phase == 0) {
        // BCE-with-logits, stable: L = log1p(exp(-|x|))
        float L = log1pf(__expf(-fabsf(xv)));
        a_bce += L + yv * fmaxf(-xv, 0.f) + (1.f - yv) * fmaxf(xv, 0.f);
        a_p  += pv;
        a_y  += yv;
        a_py += pv * yv;
        a_skp  += s[k];
        a_skpy += s[k] * yv;
      } else {
        a_skt  += s[k];
        a_sktp += s[k] * pv;
      }
    }
    __syncthreads();   // buffers reused next phase
  }

  // ---- block reduction: WMMA wave sums, then cross-wave via LDS
  float acc[8] = {a_bce, a_p, a_y, a_py, a_skp, a_skpy, a_skt, a_sktp};
  const int wave = tid >> 5;
  const int lane = tid & 31;
  for (int q = 0; q < 8; ++q) {
    float ws = wave_sum32(acc[q]);
    if (lane == 0) wred[wave * 8 + q] = ws;
  }
  __syncthreads();
  if (tid < 8) {
    float t = 0.f;
    for (int w = 0; w < 8; ++w) t += wred[w * 8 + tid];
    size_t bid = ((size_t)blockIdx.z * gridDim.y + blockIdx.y) * gridDim.x
                 + blockIdx.x;
    partial[bid * 8 + tid] = t;
  }
}

__global__ __launch_bounds__(NTHREADS)
void cldice_stage2(const float* __restrict__ partial,
                   float* __restrict__ out, int nblocks) {
  __shared__ float sm[NTHREADS][8];
  float loc[8] = {0.f, 0.f, 0.f, 0.f, 0.f, 0.f, 0.f, 0.f};
  for (int b = threadIdx.x; b < nblocks; b += NTHREADS)
    for (int q = 0; q < 8; ++q) loc[q] += partial[(size_t)b * 8 + q];
  for (int q = 0; q < 8; ++q) sm[threadIdx.x][q] = loc[q];
  __syncthreads();
  for (int off = NTHREADS / 2; off > 0; off >>= 1) {
    if (threadIdx.x < off)
      for (int q = 0; q < 8; ++q) sm[threadIdx.x][q] += sm[threadIdx.x + off][q];
    __syncthreads();
  }
  if (threadIdx.x == 0) {
    const float N = 16.0f * 1024.0f * 1024.0f;
    float bce   = sm[0][0] / N;
    float dice  = (2.0f * sm[0][3] + 1.0f) / (sm[0][1] + sm[0][2] + 1.0f);
    float tprec = (sm[0][5] + 1.0f) / (sm[0][4] + 1.0f);
    float tsens = (sm[0][7] + 1.0f) / (sm[0][6] + 1.0f);
    float cl    = 2.0f * tprec * tsens / (tprec + tsens + 1e-7f);
    out[0] = 0.3f * bce + 0.3f * (1.0f - dice) + 0.4f * (1.0f - cl);
  }
}

extern "C" void kernel_launch(void* const* d_in, const int* in_sizes, int n_in,
                              void* d_out, int out_size, void* d_ws, size_t ws_size,
                              hipStream_t stream) {
  const float* pred = (const float*)d_in[0];
  const float* targ = (const float*)d_in[1];
  float* partial = (float*)d_ws;          // 4096 blocks * 8 floats = 128 KB
  (void)in_sizes; (void)n_in; (void)out_size; (void)ws_size;

  dim3 grid(IMG / TILE, IMG / TILE, 16);  // 16 x 16 x 16 = 4096 tiles
  dim3 block(16, 16);
  cldice_stage1<<<grid, block, 0, stream>>>(pred, targ, partial);
  cldice_stage2<<<1, NTHREADS, 0, stream>>>(partial, (float*)d_out,
                                            (IMG / TILE) * (IMG / TILE) * 16);
}